// GCNLayer_32736240730563
// MI455X (gfx1250) — compile-verified
//
#include <hip/hip_runtime.h>
#include <hip/hip_bf16.h>

#define CDIM 128
#define LN_EPS 1e-5f

typedef float v2f __attribute__((ext_vector_type(2)));
typedef float v8f __attribute__((ext_vector_type(8)));

__device__ __forceinline__ float wave_sum32(float v) {
#pragma unroll
    for (int o = 16; o > 0; o >>= 1) v += __shfl_xor(v, o, 32);
    return v;
}

// ---------------------------------------------------------------------------
// init: accum = 0, deg = 1 (self loop)
// ---------------------------------------------------------------------------
__global__ void gcn_init_kernel(float* __restrict__ accum, float* __restrict__ deg, int N) {
    size_t idx = (size_t)blockIdx.x * blockDim.x + threadIdx.x;
    size_t total = (size_t)N * CDIM;
    if (idx < total) accum[idx] = 0.0f;
    if (idx < (size_t)N) deg[idx] = 1.0f;
}

// ---------------------------------------------------------------------------
// deg[dst] += 1 per edge
// ---------------------------------------------------------------------------
__global__ void gcn_degree_kernel(const int* __restrict__ ei, float* __restrict__ deg, int E) {
    int e = blockIdx.x * blockDim.x + threadIdx.x;
    if (e >= E) return;
    int dst = ei[E + e];
    unsafeAtomicAdd(&deg[dst], 1.0f);
}

// ---------------------------------------------------------------------------
// deg -> rsqrt(deg) in place (deg >= 1 always)
// ---------------------------------------------------------------------------
__global__ void gcn_dinv_kernel(float* __restrict__ deg, int N) {
    int i = blockIdx.x * blockDim.x + threadIdx.x;
    if (i < N) deg[i] = rsqrtf(deg[i]);
}

// ---------------------------------------------------------------------------
// h = x @ W^T  via V_WMMA_F32_16X16X4_F32 (full fp32 precision).
// Block = 256 threads = 8 waves; block covers 16 rows, wave w covers cols
// [16w, 16w+16). K loop = 128 in steps of 4 -> 32 WMMA per wave.
// fp32 A 16x4 layout (ISA 7.12.2): lane&15 = M row, lane>>4 selects K pair.
// fp32 B 4x16 by symmetry: lane&15 = N col, lane>>4 selects K-row pair.
// ---------------------------------------------------------------------------
__global__ void __launch_bounds__(256)
gcn_gemm_kernel(const float* __restrict__ x, const float* __restrict__ W,
                float* __restrict__ h, int N) {
    const int wave = threadIdx.x >> 5;
    const int lane = threadIdx.x & 31;
    const int half = lane >> 4;   // 0 or 1
    const int l    = lane & 15;
    const int row_base = blockIdx.x * 16;
    const int col_base = wave * 16;

    int m = row_base + l;
    if (m >= N) m = N - 1;        // clamp loads; stores guarded below (EXEC stays all-1)

    const float* xrow = x + (size_t)m * CDIM + half * 2;                   // A: x[m][k]
    const float* wrow = W + (size_t)(col_base + l) * CDIM + half * 2;      // B: W[n][k] = (W^T)[k][n]

    v8f acc = {};
#pragma unroll
    for (int k0 = 0; k0 < CDIM; k0 += 4) {
        v2f a = *(const v2f*)(xrow + k0);
        v2f b = *(const v2f*)(wrow + k0);
        acc = __builtin_amdgcn_wmma_f32_16x16x4_f32(
            /*neg_a=*/false, a, /*neg_b=*/false, b,
            /*c_mod=*/(short)0, acc, /*reuse_a=*/false, /*reuse_b=*/false);
    }

    // D layout: VGPR r -> row (r + 8*half), col = lane&15
    const int outcol = col_base + l;
#pragma unroll
    for (int r = 0; r < 8; ++r) {
        int orow = row_base + r + half * 8;
        if (orow < N) h[(size_t)orow * CDIM + outcol] = acc[r];
    }
}

// ---------------------------------------------------------------------------
// scatter: accum[dst] += h[src] * dinv[src]*dinv[dst]
// one wave per edge, 4 channels per lane (float4 gather + 4 fp32 atomics)
// h/accum are L2-resident (51MB each << 192MB L2)
// ---------------------------------------------------------------------------
__global__ void __launch_bounds__(256)
gcn_scatter_kernel(const float* __restrict__ h, const float* __restrict__ dinv,
                   const int* __restrict__ ei, float* __restrict__ accum, int E) {
    int wid = blockIdx.x * (blockDim.x >> 5) + (threadIdx.x >> 5);
    if (wid >= E) return;
    int lane = threadIdx.x & 31;

    int src = ei[wid];
    int dst = ei[E + wid];
    float norm = dinv[src] * dinv[dst];

    const float4 hv = *(const float4*)(h + (size_t)src * CDIM + lane * 4);
    float* ap = accum + (size_t)dst * CDIM + lane * 4;
    unsafeAtomicAdd(ap + 0, hv.x * norm);
    unsafeAtomicAdd(ap + 1, hv.y * norm);
    unsafeAtomicAdd(ap + 2, hv.z * norm);
    unsafeAtomicAdd(ap + 3, hv.w * norm);
}

// ---------------------------------------------------------------------------
// epilogue per node (one wave each):
//   gcn = accum + dinv^2*h (self loop) + b ; relu ; LN1 ; +x ; LN2 -> out
// ---------------------------------------------------------------------------
__global__ void __launch_bounds__(256)
gcn_epilogue_kernel(const float* __restrict__ x, const float* __restrict__ h,
                    const float* __restrict__ accum, const float* __restrict__ dinv,
                    const float* __restrict__ bvec,
                    const float* __restrict__ g1, const float* __restrict__ be1,
                    const float* __restrict__ g2, const float* __restrict__ be2,
                    float* __restrict__ out, int N) {
    int wid = blockIdx.x * (blockDim.x >> 5) + (threadIdx.x >> 5);
    if (wid >= N) return;
    int lane = threadIdx.x & 31;
    int c = lane * 4;
    size_t base = (size_t)wid * CDIM + c;

    float di = dinv[wid];
    float sw = di * di;

    float4 hv = *(const float4*)(h + base);
    float4 av = *(const float4*)(accum + base);
    float4 bv = *(const float4*)(bvec + c);

    float v0 = fmaxf(av.x + sw * hv.x + bv.x, 0.0f);
    float v1 = fmaxf(av.y + sw * hv.y + bv.y, 0.0f);
    float v2 = fmaxf(av.z + sw * hv.z + bv.z, 0.0f);
    float v3 = fmaxf(av.w + sw * hv.w + bv.w, 0.0f);

    // LayerNorm 1
    float mu = wave_sum32(v0 + v1 + v2 + v3) * (1.0f / CDIM);
    float d0 = v0 - mu, d1 = v1 - mu, d2 = v2 - mu, d3 = v3 - mu;
    float var = wave_sum32(d0 * d0 + d1 * d1 + d2 * d2 + d3 * d3) * (1.0f / CDIM);
    float inv = rsqrtf(var + LN_EPS);
    float4 g1v = *(const float4*)(g1 + c);
    float4 b1v = *(const float4*)(be1 + c);
    float4 xv  = *(const float4*)(x + base);
    float z0 = xv.x + (d0 * inv * g1v.x + b1v.x);
    float z1 = xv.y + (d1 * inv * g1v.y + b1v.y);
    float z2 = xv.z + (d2 * inv * g1v.z + b1v.z);
    float z3 = xv.w + (d3 * inv * g1v.w + b1v.w);

    // LayerNorm 2
    float mu2 = wave_sum32(z0 + z1 + z2 + z3) * (1.0f / CDIM);
    float e0 = z0 - mu2, e1 = z1 - mu2, e2 = z2 - mu2, e3 = z3 - mu2;
    float var2 = wave_sum32(e0 * e0 + e1 * e1 + e2 * e2 + e3 * e3) * (1.0f / CDIM);
    float inv2 = rsqrtf(var2 + LN_EPS);
    float4 g2v = *(const float4*)(g2 + c);
    float4 b2v = *(const float4*)(be2 + c);

    float4 o;
    o.x = e0 * inv2 * g2v.x + b2v.x;
    o.y = e1 * inv2 * g2v.y + b2v.y;
    o.z = e2 * inv2 * g2v.z + b2v.z;
    o.w = e3 * inv2 * g2v.w + b2v.w;
    *(float4*)(out + base) = o;
}

// ---------------------------------------------------------------------------
extern "C" void kernel_launch(void* const* d_in, const int* in_sizes, int n_in,
                              void* d_out, int out_size, void* d_ws, size_t ws_size,
                              hipStream_t stream) {
    const float* x  = (const float*)d_in[0];
    const int*   ei = (const int*)d_in[1];
    const float* W  = (const float*)d_in[2];
    const float* b  = (const float*)d_in[3];
    const float* g1 = (const float*)d_in[4];
    const float* be1 = (const float*)d_in[5];
    const float* g2 = (const float*)d_in[6];
    const float* be2 = (const float*)d_in[7];
    float* out = (float*)d_out;

    const int N = in_sizes[0] / CDIM;
    const int E = in_sizes[1] / 2;

    // workspace: h[N*C] | accum[N*C] | deg/dinv[N]
    float* h     = (float*)d_ws;
    float* accum = h + (size_t)N * CDIM;
    float* deg   = accum + (size_t)N * CDIM;

    const size_t total = (size_t)N * CDIM;
    gcn_init_kernel<<<(int)((total + 255) / 256), 256, 0, stream>>>(accum, deg, N);
    gcn_degree_kernel<<<(E + 255) / 256, 256, 0, stream>>>(ei, deg, E);
    gcn_dinv_kernel<<<(N + 255) / 256, 256, 0, stream>>>(deg, N);
    gcn_gemm_kernel<<<(N + 15) / 16, 256, 0, stream>>>(x, W, h, N);
    gcn_scatter_kernel<<<(E + 7) / 8, 256, 0, stream>>>(h, deg, ei, accum, E);
    gcn_epilogue_kernel<<<(N + 7) / 8, 256, 0, stream>>>(x, h, accum, deg, b,
                                                         g1, be1, g2, be2, out, N);
}